// SSIMLoss_31086973289034
// MI455X (gfx1250) — compile-verified
//
#include <hip/hip_runtime.h>
#include <hip/hip_bf16.h>

typedef __attribute__((ext_vector_type(2))) float v2f;
typedef __attribute__((ext_vector_type(8))) float v8f;

#define IMG_H 512
#define IMG_W 512
#define N_PLANES 48            // 16 batch * 3 channels
#define TILES_PER_PLANE 1024   // 32x32 tiles of 16x16
#define N_TILES (N_PLANES * TILES_PER_PLANE)
#define WAVES_PER_BLOCK 4
#define LDS_STRIDE 34          // 16 rows x 34 floats: even stride keeps 8B align
#define N_PIXELS 12582912.0    // 16*3*512*512

// Normalized 11-tap Gaussian, sigma = 1.5 (matches reference to f32 precision)
__constant__ float G11[11] = {
    0.001028381f, 0.007598758f, 0.036000770f, 0.109361200f, 0.213005510f,
    0.266011720f,
    0.213005510f, 0.109361200f, 0.036000770f, 0.007598758f, 0.001028381f};

__device__ __forceinline__ v8f wmma4(v2f a, v2f b, v8f c) {
    // D = A(16x4,f32) * B(4x16,f32) + C(16x16,f32)
    return __builtin_amdgcn_wmma_f32_16x16x4_f32(
        false, a, false, b, (short)0, c, false, false);
}

__global__ __launch_bounds__(WAVES_PER_BLOCK * 32)
void ssim_main(const float* __restrict__ pred, const float* __restrict__ targ,
               double* __restrict__ acc) {
    __shared__ float lds[WAVES_PER_BLOCK * 16 * LDS_STRIDE];

    const int tid  = threadIdx.x;
    const int lane = tid & 31;
    const int wave = tid >> 5;
    const int m    = lane & 15;     // row (A/M) or col (B/N) index within fragment
    const int half = lane >> 4;     // lane half selects K sub-pair / M+8 rows
    const int kh   = 2 * half;

    const int tileId = blockIdx.x * WAVES_PER_BLOCK + wave;
    const int plane  = tileId >> 10;
    const int t      = tileId & 1023;
    const int row0   = (t >> 5) << 4;
    const int col0   = (t & 31) << 4;

    // wave-uniform plane base offset -> SGPR base, saddr-form loads
    const int planeOff = __builtin_amdgcn_readfirstlane(plane * (IMG_H * IMG_W));
    const float* __restrict__ P = pred + planeOff;
    const float* __restrict__ T = targ + planeOff;

    // Band-weight fragments. wf[c] holds g[k - m] for k = 4c+kh, 4c+kh+1:
    // serves as A (16x4 chunk of Wv, lane=M) in the vertical pass AND as
    // B (4x16 chunk of Wh, lane=N) in the horizontal pass.
    v2f wf[7];
#pragma unroll
    for (int c = 0; c < 7; ++c) {
        const int k0 = 4 * c + kh;
        const int d0 = k0 - m, d1 = k0 + 1 - m;
        wf[c].x = ((unsigned)d0 <= 10u) ? G11[d0] : 0.0f;
        wf[c].y = ((unsigned)d1 <= 10u) ? G11[d1] : 0.0f;
    }

    // Column index / clamp / in-bounds mask: invariant across k-chunks.
    const int r0 = row0 - 5;
    const int c0 = col0 - 5;
    int  gcol[2];
    bool colin[2];
#pragma unroll
    for (int cg = 0; cg < 2; ++cg) {
        const int c = c0 + 16 * cg + m;
        colin[cg] = (unsigned)c < (unsigned)IMG_W;
        gcol[cg]  = min(max(c, 0), IMG_W - 1);   // v_med3_i32
    }

    // ---- Vertical pass: V(16x32) = Wv(16x28) x In(28x32); all 5 quantities
    // share each pair of unconditional (clamped) global loads + cndmask zero.
    v8f Vp[2] = {}, Vt[2] = {}, Vpp[2] = {}, Vtt[2] = {}, Vpt[2] = {};
#pragma unroll
    for (int c = 0; c < 7; ++c) {
        const int ra = r0 + 4 * c + kh;
        const int rb = ra + 1;
        const bool rain = (unsigned)ra < (unsigned)IMG_H;
        const bool rbin = (unsigned)rb < (unsigned)IMG_H;
        const int  rowa = min(max(ra, 0), IMG_H - 1) * IMG_W;
        const int  rowb = min(max(rb, 0), IMG_H - 1) * IMG_W;
#pragma unroll
        for (int cg = 0; cg < 2; ++cg) {
            const bool oka = rain && colin[cg];
            const bool okb = rbin && colin[cg];
            float pa = P[rowa + gcol[cg]];
            float pb = P[rowb + gcol[cg]];
            float ta = T[rowa + gcol[cg]];
            float tb = T[rowb + gcol[cg]];
            v2f bp, bt;
            bp.x = oka ? pa : 0.0f;  bp.y = okb ? pb : 0.0f;
            bt.x = oka ? ta : 0.0f;  bt.y = okb ? tb : 0.0f;
            const v2f bpp = bp * bp, btt = bt * bt, bpt = bp * bt;
            Vp[cg]  = wmma4(wf[c], bp,  Vp[cg]);
            Vt[cg]  = wmma4(wf[c], bt,  Vt[cg]);
            Vpp[cg] = wmma4(wf[c], bpp, Vpp[cg]);
            Vtt[cg] = wmma4(wf[c], btt, Vtt[cg]);
            Vpt[cg] = wmma4(wf[c], bpt, Vpt[cg]);
        }
    }

    // ---- Horizontal pass: Out(16x16) = V(16x28) x Wh(28x16).
    // V is in C/D layout; relayout to A layout through per-wave LDS.
    float* L = lds + wave * (16 * LDS_STRIDE);
    auto horiz = [&](const v8f* V) -> v8f {
#pragma unroll
        for (int cg = 0; cg < 2; ++cg)
#pragma unroll
            for (int i = 0; i < 8; ++i)
                L[(i + 8 * half) * LDS_STRIDE + 16 * cg + m] = V[cg][i];
        // in-wave LDS RAW: DS ops complete in order; wait + compiler fence
        asm volatile("s_wait_dscnt 0" ::: "memory");
        v8f O = {};
#pragma unroll
        for (int c = 0; c < 7; ++c) {
            const int k0 = 4 * c + kh;
            v2f a;
            a.x = L[m * LDS_STRIDE + k0];
            a.y = L[m * LDS_STRIDE + k0 + 1];
            O = wmma4(a, wf[c], O);
        }
        asm volatile("s_wait_dscnt 0" ::: "memory"); // WAR fence before reuse
        return O;
    };

    const v8f Omp = horiz(Vp);
    const v8f Omt = horiz(Vt);
    const v8f Opp = horiz(Vpp);
    const v8f Ott = horiz(Vtt);
    const v8f Opt = horiz(Vpt);

    // ---- Pointwise SSIM map + partial sum (8 pixels per lane)
    const float C1 = 1e-4f;   // (0.01*range)^2
    const float C2 = 9e-4f;   // (0.03*range)^2
    float s = 0.0f;
#pragma unroll
    for (int i = 0; i < 8; ++i) {
        const float mp  = Omp[i], mt = Omt[i];
        const float mpp = mp * mp, mtt = mt * mt, mpt = mp * mt;
        const float sp  = fmaxf(Opp[i] - mpp, 0.0f);
        const float st  = fmaxf(Ott[i] - mtt, 0.0f);
        const float spt = Opt[i] - mpt;
        const float num = (2.0f * mpt + C1) * (2.0f * spt + C2);
        const float den = (mpp + mtt + C1) * (sp + st + C2);
        s += num / den;
    }

    // wave32 butterfly reduce, one f64 atomic per wave
#pragma unroll
    for (int off = 16; off; off >>= 1)
        s += __shfl_xor(s, off, 32);
    if (lane == 0)
        atomicAdd(acc, (double)s);
}

__global__ void ssim_init(double* acc) { *acc = 0.0; }

__global__ void ssim_finish(const double* __restrict__ acc, float* __restrict__ out) {
    out[0] = (float)(1.0 - acc[0] * (1.0 / N_PIXELS));
}

extern "C" void kernel_launch(void* const* d_in, const int* in_sizes, int n_in,
                              void* d_out, int out_size, void* d_ws, size_t ws_size,
                              hipStream_t stream) {
    const float* pred = (const float*)d_in[0];
    const float* targ = (const float*)d_in[1];
    double* acc = (double*)d_ws;
    float* out  = (float*)d_out;

    ssim_init<<<1, 1, 0, stream>>>(acc);
    ssim_main<<<N_TILES / WAVES_PER_BLOCK, WAVES_PER_BLOCK * 32, 0, stream>>>(
        pred, targ, acc);
    ssim_finish<<<1, 1, 0, stream>>>(acc, out);
}